// Steady_4363686772947
// MI455X (gfx1250) — compile-verified
//
#include <hip/hip_runtime.h>
#include <hip/hip_bf16.h>

typedef __attribute__((ext_vector_type(2))) float v2f;
typedef __attribute__((ext_vector_type(4))) float f4v;
typedef __attribute__((ext_vector_type(8))) float v8f;

#define PI2 6.28318530717958647692f

// ---------------------------------------------------------------------------
// k0: build the y-DFT basis table bas[y][n], y=0..255, n=0..15:
//   n in 0..3  : cos(2*pi*n*y/256)      (ky = n)
//   n in 4..7  : sin(2*pi*(n-4)*y/256)
//   n in 8..15 : 0   (WMMA N-padding)
// ---------------------------------------------------------------------------
__global__ void __launch_bounds__(256) k0_basis(float* __restrict__ basg) {
    const int y = threadIdx.x;
#pragma unroll
    for (int ky = 0; ky < 4; ++ky) {
        float s, c;
        __sincosf(PI2 * (float)(ky * y) * (1.0f / 256.0f), &s, &c);
        basg[y * 16 + ky]     = c;
        basg[y * 16 + 4 + ky] = s;
    }
#pragma unroll
    for (int n = 8; n < 16; ++n) basg[y * 16 + n] = 0.0f;
}

// ---------------------------------------------------------------------------
// k1a: y-DFT via fp32 WMMA.  One wave = one 16-row tile of one image.
//   D[M,N] = sum_y xs[x=rt*16+M, y] * bas[y][N],  K accumulated in chunks of 4.
//   xs[x,y] = x_full[2x, 2y]  (the ::2,::2 downsample).
// Output G[bi][x][8]: n=0..3 cos-sums (a), n=4..7 sin-sums (b); FW1_y = a - i b.
// grid = 64 images * 8 tile-groups = 512 blocks, blockDim = 64 (2 waves).
// ---------------------------------------------------------------------------
__global__ void __launch_bounds__(64) k1a_ydft(const float* __restrict__ x,
                                               const float* __restrict__ basg,
                                               float* __restrict__ Gws) {
    __shared__ float bas[256 * 16];  // 16KB
    const int blk = blockIdx.x;
    const int bi  = blk >> 3;   // (b,i) image, 0..63
    const int tg  = blk & 7;    // tile group: tiles {2*tg, 2*tg+1}

    // stage basis into LDS (16KB, 1024 float4s)
    for (int idx = threadIdx.x; idx < 1024; idx += 64)
        ((f4v*)bas)[idx] = ((const f4v*)basg)[idx];
    __syncthreads();

    const int lane = threadIdx.x & 31;
    const int wave = threadIdx.x >> 5;   // 0..1
    const int h    = lane >> 4;          // lane half
    const int ln   = lane & 15;
    const int rt   = tg * 2 + wave;      // row tile 0..15
    const int xrow = rt * 16 + ln;       // downsampled row, 0..255

    // A frag layout (32-bit A 16x4): lane ln row M, K = v + 2*h.
    // Element A[M, 4k+K] = x_full[2*xrow][2*(4k+K)] -> float4 load at col 8k+4h,
    // keep elements 0 and 2.
    const float* rowp = x + (size_t)bi * (512 * 512) + (size_t)(2 * xrow) * 512 + 4 * h;

    v8f acc = {};
#pragma unroll 4
    for (int k = 0; k < 64; ++k) {
        __builtin_prefetch(rowp + 8 * k + 96, 0, 0);   // global_prefetch ahead
        const f4v q = __builtin_nontemporal_load((const f4v*)(rowp + 8 * k));
        v2f a;
        a.x = q.x;   // y = 4k + 2h
        a.y = q.z;   // y = 4k + 2h + 1
        const int yb = 4 * k + 2 * h;
        v2f bf;
        bf.x = bas[yb * 16 + ln];        // B[K=2h,   N=ln]
        bf.y = bas[yb * 16 + 16 + ln];   // B[K=2h+1, N=ln]
        acc = __builtin_amdgcn_wmma_f32_16x16x4_f32(
            false, a, false, bf, (short)0, acc, false, false);
    }

    // C/D layout: vgpr j -> M = j + 8h, N = lane%16.  Keep N<8 columns.
    if (ln < 8) {
        float* gp = Gws + bi * 2048 + (rt * 16 + 8 * h) * 8 + ln;
#pragma unroll
        for (int j = 0; j < 8; ++j) gp[j * 8] = acc[j];
    }
}

// ---------------------------------------------------------------------------
// k1b: x-DFT reduction: Fhat[kx,ky] = sum_x (a - i b)(cos - i sin),
//   phi = 2*pi*n*x/256, n = kidx<4 ? kidx : kidx-8.
//   re = sum(a*c - b*s),  im = -sum(a*s + b*c).
// Fh layout: [bi][kidx*8 + ky*2 + part].  grid=64, blockDim=256.
// ---------------------------------------------------------------------------
__global__ void __launch_bounds__(256) k1b_xdft(const float* __restrict__ Gws,
                                                float* __restrict__ Fh) {
    __shared__ float part[64 * 4];
    const int bi   = blockIdx.x;
    const int tid  = threadIdx.x;
    const int q    = tid >> 6;        // x-quarter
    const int oidx = tid & 63;        // kidx*8 + ky*2 + part
    const int kidx = oidx >> 3;
    const int ky   = (oidx >> 1) & 3;
    const int pr   = oidx & 1;
    const float n  = (kidx < 4) ? (float)kidx : (float)(kidx - 8);
    const float* G = Gws + bi * 2048;

    float sum = 0.0f;
    for (int xi = 0; xi < 64; ++xi) {
        const int xx  = q * 64 + xi;
        const float a = G[xx * 8 + ky];
        const float b = G[xx * 8 + 4 + ky];
        float s, c;
        __sincosf(PI2 * n * (float)xx * (1.0f / 256.0f), &s, &c);
        sum += pr ? -(a * s + b * c) : (a * c - b * s);
    }
    part[oidx * 4 + q] = sum;
    __syncthreads();
    if (tid < 64)
        Fh[bi * 64 + tid] = part[tid * 4] + part[tid * 4 + 1] +
                            part[tid * 4 + 2] + part[tid * 4 + 3];
}

// ---------------------------------------------------------------------------
// k3: fused mixing + synthesis.  grid = 64 (b,o) * 8 z-tiles = 512 blocks.
//   out2[kidx,ky] = sum_i Fhat[b,i,kidx,ky] * w[i,o]
//   A[z,ky]       = sum_kidx out2 * exp(+2*pi*i*n*z/512)
//   out[z,t]      = (1/65536) * Re sum_ky A[z,ky] * exp(+2*pi*i*ky*t/512)
// ky=0 phase is 1 -> folded into per-row constant; only ky=1..3 tabulated.
// ---------------------------------------------------------------------------
__global__ void __launch_bounds__(256) k3_synth(const float* __restrict__ Fh,
                                                const float* __restrict__ w1,
                                                const float* __restrict__ w2,
                                                float* __restrict__ out) {
    __shared__ float e2[6 * 512];   // [ (ky-1)*2 + {c,s} ][ t ], 12KB
    __shared__ float Az[64 * 8];    // [zloc][ky*2 + {re,im}]
    __shared__ float o2[64];        // [ (kidx*4+ky)*2 + {re,im} ]

    const int blk = blockIdx.x;
    const int bo  = blk >> 3;       // b*8 + o
    const int zt  = blk & 7;        // z tile (64 rows each)
    const int b   = bo >> 3;
    const int o   = bo & 7;
    const int tid = threadIdx.x;

    // out2: 32 complex modes
    if (tid < 32) {
        const int kidx = tid >> 2, ky = tid & 3;
        const float* wsrc = (kidx < 4) ? w1 : w2;
        const int mx = kidx & 3;
        float re = 0.0f, im = 0.0f;
#pragma unroll
        for (int i = 0; i < 8; ++i) {
            const float Fr = Fh[(b * 8 + i) * 64 + tid * 2];
            const float Fi = Fh[(b * 8 + i) * 64 + tid * 2 + 1];
            const int widx = (((i * 8 + o) * 4 + mx) * 4 + ky) * 2;
            const float wr = wsrc[widx], wi = wsrc[widx + 1];
            re += Fr * wr - Fi * wi;
            im += Fr * wi + Fi * wr;
        }
        o2[tid * 2]     = re;
        o2[tid * 2 + 1] = im;
    }

    // E2 table for ky = 1..3
    for (int t = tid; t < 512; t += 256) {
#pragma unroll
        for (int m = 1; m <= 3; ++m) {
            float s, c;
            __sincosf(PI2 * (float)(m * t) * (1.0f / 512.0f), &s, &c);
            e2[(2 * (m - 1)) * 512 + t]     = c;
            e2[(2 * (m - 1) + 1) * 512 + t] = s;
        }
    }
    __syncthreads();

    // A[z,ky] for this block's 64 z rows (256 threads = 64 rows * 4 ky)
    {
        const int zloc = tid >> 2, ky = tid & 3;
        const int z = zt * 64 + zloc;
        float re = 0.0f, im = 0.0f;
#pragma unroll
        for (int kidx = 0; kidx < 8; ++kidx) {
            const float n = (kidx < 4) ? (float)kidx : (float)(kidx - 8);
            float s, c;
            __sincosf(PI2 * n * (float)z * (1.0f / 512.0f), &s, &c);
            const float xr = o2[(kidx * 4 + ky) * 2];
            const float xi = o2[(kidx * 4 + ky) * 2 + 1];
            re += xr * c - xi * s;
            im += xr * s + xi * c;
        }
        Az[zloc * 8 + ky * 2]     = re;
        Az[zloc * 8 + ky * 2 + 1] = im;
    }
    __syncthreads();

    // Stream the output: 64 rows * 512 t, nontemporal stores.
    const float inv = 1.0f / 65536.0f;
    float* obase = out + ((size_t)bo * 512 + (size_t)zt * 64) * 512;
    for (int zloc = 0; zloc < 64; ++zloc) {
        const float ar0 = Az[zloc * 8 + 0];                              // ky=0: e2==1
        const float ar1 = Az[zloc * 8 + 2], ai1 = Az[zloc * 8 + 3];
        const float ar2 = Az[zloc * 8 + 4], ai2 = Az[zloc * 8 + 5];
        const float ar3 = Az[zloc * 8 + 6], ai3 = Az[zloc * 8 + 7];
        float* orow = obase + (size_t)zloc * 512;
#pragma unroll
        for (int j = 0; j < 2; ++j) {
            const int t = j * 256 + tid;
            const float c1 = e2[0 * 512 + t], s1 = e2[1 * 512 + t];
            const float c2 = e2[2 * 512 + t], s2 = e2[3 * 512 + t];
            const float c3 = e2[4 * 512 + t], s3 = e2[5 * 512 + t];
            const float v = ar0 + ar1 * c1 - ai1 * s1
                                + ar2 * c2 - ai2 * s2
                                + ar3 * c3 - ai3 * s3;
            __builtin_nontemporal_store(v * inv, orow + t);
        }
    }
}

// ---------------------------------------------------------------------------
// host launcher
// inputs: 0=x (8*8*512*512 f32), 1=T (512), 2=X (512), 3=weights1_ri (2048),
//         4=weights2_ri (2048).  out: 8*8*512*512 f32.
// ws layout (floats): [0,4096)=basis, [4096,135168)=G, [135168,139264)=Fhat.
// ---------------------------------------------------------------------------
extern "C" void kernel_launch(void* const* d_in, const int* in_sizes, int n_in,
                              void* d_out, int out_size, void* d_ws, size_t ws_size,
                              hipStream_t stream) {
    const float* x  = (const float*)d_in[0];
    const float* w1 = (const float*)d_in[3];
    const float* w2 = (const float*)d_in[4];
    float* outp = (float*)d_out;
    float* ws   = (float*)d_ws;

    float* bas = ws;                   // 4096 floats
    float* Gws = ws + 4096;            // 64*256*8 = 131072 floats
    float* Fh  = ws + 4096 + 131072;   // 64*64   = 4096 floats

    k0_basis<<<1, 256, 0, stream>>>(bas);
    k1a_ydft<<<512, 64, 0, stream>>>(x, bas, Gws);
    k1b_xdft<<<64, 256, 0, stream>>>(Gws, Fh);
    k3_synth<<<512, 256, 0, stream>>>(Fh, w1, w2, outp);
}